// CapsuleNetwork_11596411699769
// MI455X (gfx1250) — compile-verified
//
#include <hip/hip_runtime.h>
#include <stdint.h>

typedef __attribute__((ext_vector_type(2))) float v2f;
typedef __attribute__((ext_vector_type(8))) float v8f;

// ---------------------------------------------------------------------------
// CDNA5 async memory->LDS copy (ASYNCcnt-tracked), per ISA 10.x / 08_async.
// lds_off: byte offset within the wave's LDS allocation (low 32 bits of the
// generic address of a __shared__ object). gptr: per-lane global address.
// ---------------------------------------------------------------------------
__device__ __forceinline__ void async_ld_lds_b32(uint32_t lds_off,
                                                 const float* gptr) {
  asm volatile("global_load_async_to_lds_b32 %0, %1, off"
               :: "v"(lds_off), "v"(gptr)
               : "memory");
}
__device__ __forceinline__ void wait_asynccnt0() {
  asm volatile("s_wait_asynccnt 0x0" ::: "memory");
}

// ---------------------------------------------------------------------------
// conv1 + ReLU: x[256,1,28,28] -> h[256,256,20,20]
// grid (B=256, 16 co-groups), block 256. Image + 16 filters staged into LDS
// with async-to-LDS DMA instead of load/ds_store pairs.
// ---------------------------------------------------------------------------
__global__ void conv1_relu_kernel(const float* __restrict__ x,
                                  const float* __restrict__ w,
                                  const float* __restrict__ bias,
                                  float* __restrict__ h) {
  __shared__ float xs[784];
  __shared__ float wsh[16 * 81];
  const int b = blockIdx.x;
  const int cog = blockIdx.y;
  const int t = threadIdx.x;
  const float* xb = x + b * 784;
  const float* wg = w + cog * 16 * 81;
  for (int idx = t; idx < 784; idx += 256)
    async_ld_lds_b32((uint32_t)(uintptr_t)&xs[idx], xb + idx);
  for (int idx = t; idx < 16 * 81; idx += 256)
    async_ld_lds_b32((uint32_t)(uintptr_t)&wsh[idx], wg + idx);
  wait_asynccnt0();
  __syncthreads();
  for (int idx = t; idx < 16 * 400; idx += 256) {
    const int col = idx / 400;      // local out-channel
    const int sp  = idx % 400;
    const int oh = sp / 20, ow = sp % 20;
    const int co = cog * 16 + col;
    float acc = bias[co];
    const float* wl = &wsh[col * 81];
    const float* xw = &xs[oh * 28 + ow];
#pragma unroll
    for (int kh = 0; kh < 9; ++kh)
#pragma unroll
      for (int kw = 0; kw < 9; ++kw)
        acc = fmaf(wl[kh * 9 + kw], xw[kh * 28 + kw], acc);
    h[((b * 256 + co) * 20 + oh) * 20 + ow] = fmaxf(acc, 0.0f);
  }
}

// ---------------------------------------------------------------------------
// Primary-capsule conv as implicit GEMM with fp32 WMMA.
// C[M=256][N=9216] = W[256][20736] x im2col(h)[20736][9216], N = b*36+oh*6+ow.
// Each wave owns a 64x16 C tile (4 M-tiles): the scattered im2col B fragment
// is gathered ONCE per K-step and fed to 4 back-to-back
// v_wmma_f32_16x16x4_f32, while the 4 A fragments are contiguous b64 loads at
// immediate offsets. 2304 waves total.
// ---------------------------------------------------------------------------
__global__ void prim_conv_wmma(const float* __restrict__ h,
                               const float* __restrict__ w,     // [256][20736]
                               const float* __restrict__ bias,
                               float* __restrict__ p) {         // [256][256][36]
  const int lane = threadIdx.x & 31;
  const int wave = threadIdx.x >> 5;
  const int gw = blockIdx.x * 8 + wave;   // 0..2303 tiles
  const int mt4 = gw / 576;               // 4 M-quad tiles (64 rows each)
  const int nt  = gw % 576;               // 576 N tiles
  const int l16 = lane & 15;
  const int khi = (lane >> 4) * 2;        // this lane's K sub-offset (0 or 2)

  // A: 4 weight rows (m, m+16, m+32, m+48), K-contiguous
  const int m = mt4 * 64 + l16;
  const float* aptr = w + (size_t)m * 20736 + khi;

  // B: im2col gather column n
  const int n = nt * 16 + l16;
  const int bimg = n / 36;
  const int sp = n % 36;
  const int oh = sp / 6, ow = sp % 6;
  const float* hbase = h + (size_t)bimg * 102400 + (oh * 2) * 20 + ow * 2;

  // incremental K decode: off = ci*400 + kh*20 + kw, stepping K by 4
  int kw = khi, kh = 0, off = khi;

  v8f acc[4];
#pragma unroll
  for (int q = 0; q < 4; ++q)
    acc[q] = (v8f){0.f, 0.f, 0.f, 0.f, 0.f, 0.f, 0.f, 0.f};

  for (int k = 0; k < 20736; k += 4) {
    // B fragment: one gather shared by 4 WMMAs
    int off2;
    if (kw + 1 < 9)      off2 = off + 1;    // same kernel row
    else if (kh + 1 < 9) off2 = off + 12;   // next kernel row
    else                 off2 = off + 232;  // next input channel
    v2f bf;
    bf.x = hbase[off];
    bf.y = hbase[off2];

    // A fragments: contiguous streams at fixed 16-row strides
    v2f a0 = *(const v2f*)(aptr);
    v2f a1 = *(const v2f*)(aptr + 16 * 20736);
    v2f a2 = *(const v2f*)(aptr + 32 * 20736);
    v2f a3 = *(const v2f*)(aptr + 48 * 20736);
    aptr += 4;

    acc[0] = __builtin_amdgcn_wmma_f32_16x16x4_f32(false, a0, false, bf,
                                                   (short)0, acc[0], false, false);
    acc[1] = __builtin_amdgcn_wmma_f32_16x16x4_f32(false, a1, false, bf,
                                                   (short)0, acc[1], false, false);
    acc[2] = __builtin_amdgcn_wmma_f32_16x16x4_f32(false, a2, false, bf,
                                                   (short)0, acc[2], false, false);
    acc[3] = __builtin_amdgcn_wmma_f32_16x16x4_f32(false, a3, false, bf,
                                                   (short)0, acc[3], false, false);

    // advance this lane's K by 4
    kw += 4; off += 4;
    if (kw >= 9) { kw -= 9; off += 11; kh += 1;
                   if (kh >= 9) { kh -= 9; off += 220; } }
  }
  // D layout: row = r + (lane>=16)*8, col = lane&15
#pragma unroll
  for (int q = 0; q < 4; ++q) {
#pragma unroll
    for (int r = 0; r < 8; ++r) {
      const int co = mt4 * 64 + q * 16 + r + ((lane >> 4) << 3);
      p[(size_t)bimg * 9216 + co * 36 + sp] = acc[q][r] + bias[co];
    }
  }
}

// ---------------------------------------------------------------------------
// squash over the 1152-axis of u=[B,8,1152]; p[b, g*1152 .. ] is contiguous.
// Writes transposed xp[B,1152,8] for the routing GEMM.
// ---------------------------------------------------------------------------
__global__ void primary_squash(const float* __restrict__ p,
                               float* __restrict__ xp) {
  const int b = blockIdx.x, g = blockIdx.y, t = threadIdx.x;
  const float* pg = p + (size_t)b * 9216 + g * 1152;
  __shared__ float red[256];
  float acc = 0.f;
  for (int e = t; e < 1152; e += 256) { float v = pg[e]; acc = fmaf(v, v, acc); }
  red[t] = acc; __syncthreads();
  for (int s = 128; s > 0; s >>= 1) { if (t < s) red[t] += red[t + s]; __syncthreads(); }
  const float msq = red[0];
  const float f = (msq / (1.0f + msq)) * rsqrtf(msq);
  for (int e = t; e < 1152; e += 256)
    xp[((size_t)b * 1152 + e) * 8 + g] = pg[e] * f;
}

// ---------------------------------------------------------------------------
// u_hat[b,i,j,o] = sum_u W[i,j,o,u] * xp[b,i,u]
// Per i: [160x8] x [8x256] GEMM; M tile == one j, N tile over batch. fp32 WMMA.
// ---------------------------------------------------------------------------
__global__ void uhat_wmma(const float* __restrict__ Wr,   // [1152][10][16][8]
                          const float* __restrict__ xp,   // [256][1152][8]
                          float* __restrict__ uhat) {     // [256][1152][10][16]
  const int lane = threadIdx.x & 31;
  const int wave = threadIdx.x >> 5;
  const int gw = blockIdx.x * 8 + wave;   // 0 .. 1152*160-1
  const int i = gw / 160;
  const int rem = gw % 160;
  const int j = rem >> 4;                 // 10 M tiles == j
  const int nt = rem & 15;                // 16 N tiles over batch
  const int l16 = lane & 15;
  const int khi = (lane >> 4) * 2;

  const float* aptr = Wr + (((size_t)i * 10 + j) * 16 + l16) * 8 + khi;
  const int bcol = nt * 16 + l16;
  const float* bptr = xp + ((size_t)bcol * 1152 + i) * 8 + khi;

  v8f acc = {0.f, 0.f, 0.f, 0.f, 0.f, 0.f, 0.f, 0.f};
#pragma unroll
  for (int k = 0; k < 8; k += 4) {
    v2f a = *(const v2f*)(aptr + k);
    v2f bb = *(const v2f*)(bptr + k);
    acc = __builtin_amdgcn_wmma_f32_16x16x4_f32(false, a, false, bb,
                                                (short)0, acc, false, false);
  }
#pragma unroll
  for (int r = 0; r < 8; ++r) {
    const int o = r + ((lane >> 4) << 3);
    uhat[(((size_t)bcol * 1152 + i) * 10 + j) * 16 + o] = acc[r];
  }
}

// ---------------------------------------------------------------------------
// routing helpers
// ---------------------------------------------------------------------------
__global__ void zero_kernel(float* __restrict__ b, int n) {
  const int idx = blockIdx.x * 256 + threadIdx.x;
  if (idx < n) b[idx] = 0.f;
}

// softmax over axis 0 (the 1152 axis) of b_ij[1152,10]
__global__ void softmax_col(const float* __restrict__ bij,
                            float* __restrict__ cij) {
  const int j = blockIdx.x, t = threadIdx.x;
  __shared__ float red[256];
  float m = -3.4e38f;
  for (int i = t; i < 1152; i += 256) m = fmaxf(m, bij[i * 10 + j]);
  red[t] = m; __syncthreads();
  for (int s = 128; s > 0; s >>= 1) { if (t < s) red[t] = fmaxf(red[t], red[t + s]); __syncthreads(); }
  m = red[0]; __syncthreads();
  float acc = 0.f;
  for (int i = t; i < 1152; i += 256) acc += __expf(bij[i * 10 + j] - m);
  red[t] = acc; __syncthreads();
  for (int s = 128; s > 0; s >>= 1) { if (t < s) red[t] += red[t + s]; __syncthreads(); }
  const float inv = 1.0f / red[0];
  for (int i = t; i < 1152; i += 256)
    cij[i * 10 + j] = __expf(bij[i * 10 + j] - m) * inv;
}

// s[b,j,o] = sum_i c[i,j] * uhat[b,i,j,o]
__global__ void s_j_kernel(const float* __restrict__ cij,
                           const float* __restrict__ uhat,
                           float* __restrict__ s) {
  const int j = blockIdx.x, b = blockIdx.y, t = threadIdx.x;
  float acc[16];
#pragma unroll
  for (int o = 0; o < 16; ++o) acc[o] = 0.f;
  for (int i = t; i < 1152; i += 256) {
    const float c = cij[i * 10 + j];
    const float* u = uhat + (((size_t)b * 1152 + i) * 10 + j) * 16;
    // next row for this thread is 164 KB away; software prefetch it
    __builtin_prefetch(u + (size_t)256 * 160, 0, 1);
#pragma unroll
    for (int o = 0; o < 16; ++o) acc[o] = fmaf(c, u[o], acc[o]);
  }
  __shared__ float red[256 * 16];   // 16 KB of the WGP's 320 KB LDS
#pragma unroll
  for (int o = 0; o < 16; ++o) red[o * 256 + t] = acc[o];
  __syncthreads();
  for (int st = 128; st > 0; st >>= 1) {
    if (t < st) {
#pragma unroll
      for (int o = 0; o < 16; ++o) red[o * 256 + t] += red[o * 256 + t + st];
    }
    __syncthreads();
  }
  if (t < 16) s[((size_t)b * 10 + j) * 16 + t] = red[t * 256];
}

// v = squash(s) over the j(=10) axis, per (b, o) — matches the reference quirk.
__global__ void squash_v_kernel(const float* __restrict__ s,
                                float* __restrict__ v) {
  const int g = blockIdx.x * 256 + threadIdx.x;  // 4096 (b,o) pairs
  if (g >= 256 * 16) return;
  const int b = g >> 4, o = g & 15;
  float msq = 0.f;
#pragma unroll
  for (int j = 0; j < 10; ++j) { const float sv = s[(b * 10 + j) * 16 + o]; msq = fmaf(sv, sv, msq); }
  const float f = (msq / (1.0f + msq)) * rsqrtf(msq);
#pragma unroll
  for (int j = 0; j < 10; ++j) v[(b * 10 + j) * 16 + o] = s[(b * 10 + j) * 16 + o] * f;
}

// b_ij[i,j] += mean_b dot16(uhat[b,i,j,:], v[b,j,:])
__global__ void agreement_kernel(const float* __restrict__ uhat,
                                 const float* __restrict__ v,
                                 float* __restrict__ bij) {
  const int ij = blockIdx.x;          // 11520
  const int i = ij / 10, j = ij % 10;
  const int b = threadIdx.x;          // 256 == batch
  const float* u = uhat + (((size_t)b * 1152 + i) * 10 + j) * 16;
  const float* vv = v + ((size_t)b * 10 + j) * 16;
  float d = 0.f;
#pragma unroll
  for (int o = 0; o < 16; ++o) d = fmaf(u[o], vv[o], d);
  __shared__ float red[256];
  red[b] = d; __syncthreads();
  for (int s2 = 128; s2 > 0; s2 >>= 1) { if (b < s2) red[b] += red[b + s2]; __syncthreads(); }
  if (b == 0) bij[ij] += red[0] * (1.0f / 256.0f);
}

// ---------------------------------------------------------------------------
extern "C" void kernel_launch(void* const* d_in, const int* in_sizes, int n_in,
                              void* d_out, int out_size, void* d_ws, size_t ws_size,
                              hipStream_t stream) {
  (void)in_sizes; (void)n_in; (void)out_size; (void)ws_size;
  const float* x       = (const float*)d_in[0];
  const float* conv1_w = (const float*)d_in[1];
  const float* conv1_b = (const float*)d_in[2];
  const float* prim_w  = (const float*)d_in[3];
  const float* prim_b  = (const float*)d_in[4];
  const float* W_route = (const float*)d_in[5];

  char* ws = (char*)d_ws;
  size_t off = 0;
  auto alloc = [&](size_t bytes) -> void* {
    void* p0 = ws + off;
    off += (bytes + 255) & ~(size_t)255;
    return p0;
  };
  // Region 0: h (104 MB) during conv phase, reused for u_hat (189 MB) after.
  float* r0   = (float*)alloc((size_t)256 * 1152 * 10 * 16 * sizeof(float));
  float* p    = (float*)alloc((size_t)256 * 9216 * sizeof(float));
  float* xp   = (float*)alloc((size_t)256 * 1152 * 8 * sizeof(float));
  float* bij  = (float*)alloc(11520 * sizeof(float));
  float* cij  = (float*)alloc(11520 * sizeof(float));
  float* sj   = (float*)alloc(40960 * sizeof(float));
  float* vbuf = (float*)alloc(40960 * sizeof(float));
  float* h    = r0;
  float* uhat = r0;
  float* vout = (float*)d_out;

  conv1_relu_kernel<<<dim3(256, 16), 256, 0, stream>>>(x, conv1_w, conv1_b, h);
  prim_conv_wmma<<<288, 256, 0, stream>>>(h, prim_w, prim_b, p);         // 2304 waves
  primary_squash<<<dim3(256, 8), 256, 0, stream>>>(p, xp);
  uhat_wmma<<<23040, 256, 0, stream>>>(W_route, xp, uhat);               // h now dead
  zero_kernel<<<(11520 + 255) / 256, 256, 0, stream>>>(bij, 11520);

  for (int it = 0; it < 3; ++it) {
    softmax_col<<<10, 256, 0, stream>>>(bij, cij);
    s_j_kernel<<<dim3(10, 256), 256, 0, stream>>>(cij, uhat, sj);
    float* vdst = (it == 2) ? vout : vbuf;
    squash_v_kernel<<<16, 256, 0, stream>>>(sj, vdst);
    if (it < 2) agreement_kernel<<<11520, 256, 0, stream>>>(uhat, vbuf, bij);
  }
}